// MultiHeadAttention_10368051052723
// MI455X (gfx1250) — compile-verified
//
#include <hip/hip_runtime.h>

typedef float v2f __attribute__((ext_vector_type(2)));
typedef float v8f __attribute__((ext_vector_type(8)));

#define BB 2
#define SS 2048
#define DIMM 512
#define HH 8
#define DHEAD 64

__device__ __forceinline__ float neg_inf() { return -__builtin_inff(); }

// ---------------------------------------------------------------------------
// Kernel 1: padding masks  qmask[b,s] = sign(sum|queries[b,s,:]|), kmask likewise
// ---------------------------------------------------------------------------
__global__ __launch_bounds__(256) void masks_kernel(
    const float* __restrict__ queries, const float* __restrict__ keys,
    float* __restrict__ qmask, float* __restrict__ kmask)
{
    int i = blockIdx.x * blockDim.x + threadIdx.x;     // 0 .. 2*BB*SS-1
    int which = i / (BB * SS);
    int idx   = i % (BB * SS);
    const float* p = (which ? keys : queries) + (size_t)idx * DIMM;
    float s = 0.f;
    for (int k = 0; k < DIMM; ++k) s += fabsf(p[k]);
    (which ? kmask : qmask)[idx] = (s > 0.f) ? 1.f : 0.f;
}

// ---------------------------------------------------------------------------
// Kernel 2: fused QKV projection  Y = X @ W^T + b   (grid.z: 0=Q, 1=K, 2=V)
// 32x32 output per wave (2x2 register-blocked 16x16 WMMA tiles).
// A layout (16x4 f32): lane&15 = M row; VGPR v holds K = v + 2*(lane>>4).
// B layout (4x16 f32): lane&15 = N col; VGPR v holds K = v + 2*(lane>>4).
// C/D layout: acc[j] is (M = j + 8*(lane>>4), N = lane&15).
// ---------------------------------------------------------------------------
__global__ __launch_bounds__(256) void qkv_proj_kernel(
    const float* __restrict__ queries, const float* __restrict__ keys,
    const float* __restrict__ Wq, const float* __restrict__ bq,
    const float* __restrict__ Wk, const float* __restrict__ bk,
    const float* __restrict__ Wv, const float* __restrict__ bv,
    float* __restrict__ Qout, float* __restrict__ Kout, float* __restrict__ Vout)
{
    const int which = blockIdx.z;
    const float* X    = (which == 0) ? queries : keys;
    const float* W    = (which == 0) ? Wq : (which == 1) ? Wk : Wv;
    const float* bias = (which == 0) ? bq : (which == 1) ? bk : bv;
    float*       Y    = (which == 0) ? Qout : (which == 1) ? Kout : Vout;

    const int lane = threadIdx.x & 31;
    const int wave = threadIdx.x >> 5;
    const int tile = blockIdx.x * 8 + wave;   // 2048 tiles: 128 M32 x 16 N32
    const int tM = tile >> 4;
    const int tN = tile & 15;
    const int half = lane >> 4;
    const int l15  = lane & 15;

    const float* arow[2];
    const float* brow[2];
#pragma unroll
    for (int mi = 0; mi < 2; ++mi)
        arow[mi] = X + (size_t)(tM * 32 + mi * 16 + l15) * DIMM;
#pragma unroll
    for (int ni = 0; ni < 2; ++ni)
        brow[ni] = W + (size_t)(tN * 32 + ni * 16 + l15) * DIMM;

    v8f acc[2][2] = {};
    for (int k0 = 0; k0 < DIMM; k0 += 4) {
        v2f a[2], bf[2];
#pragma unroll
        for (int mi = 0; mi < 2; ++mi) {
            a[mi].x = arow[mi][k0 + 2 * half];
            a[mi].y = arow[mi][k0 + 2 * half + 1];
        }
#pragma unroll
        for (int ni = 0; ni < 2; ++ni) {
            bf[ni].x = brow[ni][k0 + 2 * half];
            bf[ni].y = brow[ni][k0 + 2 * half + 1];
        }
#pragma unroll
        for (int mi = 0; mi < 2; ++mi)
#pragma unroll
            for (int ni = 0; ni < 2; ++ni)
                acc[mi][ni] = __builtin_amdgcn_wmma_f32_16x16x4_f32(
                    false, a[mi], false, bf[ni], (short)0, acc[mi][ni], false, false);
    }
#pragma unroll
    for (int ni = 0; ni < 2; ++ni) {
        const float bb = bias[tN * 32 + ni * 16 + l15];
#pragma unroll
        for (int mi = 0; mi < 2; ++mi) {
            float* yb = Y + (size_t)(tM * 32 + mi * 16 + 8 * half) * DIMM
                          + tN * 32 + ni * 16 + l15;
#pragma unroll
            for (int j = 0; j < 8; ++j)
                yb[(size_t)j * DIMM] = acc[mi][ni][j] + bb;
        }
    }
}

// ---------------------------------------------------------------------------
// Kernel 3: scores = Q_ K_^T / sqrt(64) -> attention output region.
// 64x64 output per wave (4x4 register-blocked): 16 wmma per 8 operand loads.
// Causal tiles with kt > qt skipped wave-uniformly (softmax zero-fills them).
// ---------------------------------------------------------------------------
__global__ __launch_bounds__(256) void scores_kernel(
    const float* __restrict__ Q, const float* __restrict__ Km,
    float* __restrict__ attn, const int* __restrict__ causality)
{
    const int causal = causality[0];
    const int z = blockIdx.z;                 // b*H + h
    const int b = z / HH, h = z % HH;
    const int lane = threadIdx.x & 31;
    const int wave = threadIdx.x >> 5;
    const int t  = blockIdx.x * 8 + wave;     // 0..1023 (32x32 64-tiles)
    const int qt = t >> 5;
    const int kt = t & 31;
    if (causal && kt > qt) return;            // wave-uniform: EXEC stays full below

    const int half = lane >> 4;
    const int l15  = lane & 15;

    const float* arow[4];
    const float* brow[4];
#pragma unroll
    for (int mi = 0; mi < 4; ++mi)
        arow[mi] = Q  + (size_t)(b * SS + qt * 64 + mi * 16 + l15) * DIMM + h * DHEAD;
#pragma unroll
    for (int ni = 0; ni < 4; ++ni)
        brow[ni] = Km + (size_t)(b * SS + kt * 64 + ni * 16 + l15) * DIMM + h * DHEAD;

    v8f acc[4][4] = {};
#pragma unroll
    for (int k0 = 0; k0 < DHEAD; k0 += 4) {
        v2f a[4], bf[4];
#pragma unroll
        for (int mi = 0; mi < 4; ++mi) {
            a[mi].x = arow[mi][k0 + 2 * half];
            a[mi].y = arow[mi][k0 + 2 * half + 1];
        }
#pragma unroll
        for (int ni = 0; ni < 4; ++ni) {
            bf[ni].x = brow[ni][k0 + 2 * half];
            bf[ni].y = brow[ni][k0 + 2 * half + 1];
        }
#pragma unroll
        for (int mi = 0; mi < 4; ++mi)
#pragma unroll
            for (int ni = 0; ni < 4; ++ni)
                acc[mi][ni] = __builtin_amdgcn_wmma_f32_16x16x4_f32(
                    false, a[mi], false, bf[ni], (short)0, acc[mi][ni], false, false);
    }
    const float scale = 0.125f;               // 1/sqrt(64)
#pragma unroll
    for (int mi = 0; mi < 4; ++mi) {
#pragma unroll
        for (int ni = 0; ni < 4; ++ni) {
            float* ob = attn + ((size_t)z * SS + qt * 64 + mi * 16 + 8 * half) * SS
                             + kt * 64 + ni * 16 + l15;
#pragma unroll
            for (int j = 0; j < 8; ++j)
                ob[(size_t)j * SS] = acc[mi][ni][j] * scale;
        }
    }
}

// ---------------------------------------------------------------------------
// Kernel 4: row softmax with key mask, nan_to_num, query mask, causal zero-fill.
// One 256-thread block per row (16*2048 rows).
// ---------------------------------------------------------------------------
__global__ __launch_bounds__(256) void softmax_kernel(
    float* __restrict__ attn,
    const float* __restrict__ qmask, const float* __restrict__ kmask,
    const int* __restrict__ causality)
{
    const int causal = causality[0];
    const int row = blockIdx.x;               // z*S + q
    const int z = row >> 11;
    const int q = row & (SS - 1);
    const int b = z / HH;
    float* p = attn + (size_t)row * SS;
    const int L = causal ? (q + 1) : SS;
    const float* km = kmask + b * SS;

    __shared__ float red_max[8];
    __shared__ float red_sum[8];
    const int tid  = threadIdx.x;
    const int lane = tid & 31;
    const int wv   = tid >> 5;
    const float NI = neg_inf();

    // pass 1: masked max
    float mx = NI;
    for (int k = tid; k < L; k += 256) {
        float v = (km[k] != 0.f) ? p[k] : NI;
        mx = fmaxf(mx, v);
    }
    for (int off = 16; off > 0; off >>= 1)
        mx = fmaxf(mx, __shfl_down(mx, off, 32));
    if (lane == 0) red_max[wv] = mx;
    __syncthreads();
    if (wv == 0) {
        float v = (lane < 8) ? red_max[lane] : NI;
        for (int off = 4; off > 0; off >>= 1)
            v = fmaxf(v, __shfl_down(v, off, 32));
        if (lane == 0) red_max[0] = v;
    }
    __syncthreads();
    mx = red_max[0];
    const bool anyvalid = (mx > NI);

    // pass 2: exp + sum (store unnormalized exp)
    float sum = 0.f;
    for (int k = tid; k < L; k += 256) {
        float v = 0.f;
        if (anyvalid && km[k] != 0.f) v = __expf(p[k] - mx);
        p[k] = v;
        sum += v;
    }
    for (int off = 16; off > 0; off >>= 1)
        sum += __shfl_down(sum, off, 32);
    if (lane == 0) red_sum[wv] = sum;
    __syncthreads();
    if (wv == 0) {
        float v = (lane < 8) ? red_sum[lane] : 0.f;
        for (int off = 4; off > 0; off >>= 1)
            v += __shfl_down(v, off, 32);
        if (lane == 0) red_sum[0] = v;
    }
    __syncthreads();
    sum = red_sum[0];

    const float qm  = qmask[b * SS + q];
    const float inv = (sum > 0.f) ? (qm / sum) : 0.f;   // nan_to_num: all-masked -> 0
    for (int k = tid; k < L; k += 256) p[k] *= inv;
    for (int k = L + tid; k < SS; k += 256) p[k] = 0.f; // causal upper triangle
}

// ---------------------------------------------------------------------------
// Kernel 5: out = attn @ V_, heads merged back: out[b,q,h*64+j].
// 16x64 output per wave (1x4 accumulators): the attn fragment (dominant
// 268 MB stream) is loaded ONCE and reused for all 4 V tiles.
// Causal clips K-loop to 4*(qt+1) steps.
// ---------------------------------------------------------------------------
__global__ __launch_bounds__(256) void av_kernel(
    const float* __restrict__ attn, const float* __restrict__ V,
    float* __restrict__ out, const int* __restrict__ causality)
{
    const int causal = causality[0];
    const int lane = threadIdx.x & 31;
    const int wave = threadIdx.x >> 5;
    const int tile = blockIdx.x * 8 + wave;   // 16 z * 128 qt = 2048
    const int z  = tile >> 7;
    const int qt = tile & 127;
    const int b = z / HH, h = z % HH;
    const int half = lane >> 4;
    const int l15  = lane & 15;

    const float* arow = attn + ((size_t)z * SS + qt * 16 + l15) * SS;
    const float* bbase[4];
#pragma unroll
    for (int ni = 0; ni < 4; ++ni)
        bbase[ni] = V + (size_t)b * SS * DIMM + h * DHEAD + ni * 16 + l15;

    const int ksteps = causal ? 4 * (qt + 1) : (SS / 4);
    v8f acc[4] = {};
    for (int ks = 0; ks < ksteps; ++ks) {
        const int k0 = ks * 4;
        v2f a, bf[4];
        a.x = arow[k0 + 2 * half];
        a.y = arow[k0 + 2 * half + 1];
#pragma unroll
        for (int ni = 0; ni < 4; ++ni) {
            bf[ni].x = bbase[ni][(size_t)(k0 + 2 * half) * DIMM];
            bf[ni].y = bbase[ni][(size_t)(k0 + 2 * half + 1) * DIMM];
        }
#pragma unroll
        for (int ni = 0; ni < 4; ++ni)
            acc[ni] = __builtin_amdgcn_wmma_f32_16x16x4_f32(
                false, a, false, bf[ni], (short)0, acc[ni], false, false);
    }
#pragma unroll
    for (int ni = 0; ni < 4; ++ni) {
        float* ob = out + ((size_t)b * SS + qt * 16 + 8 * half) * DIMM
                        + h * DHEAD + ni * 16 + l15;
#pragma unroll
        for (int j = 0; j < 8; ++j)
            ob[(size_t)j * DIMM] = acc[ni][j];
    }
}

// ---------------------------------------------------------------------------
// Host-side launch
// ---------------------------------------------------------------------------
extern "C" void kernel_launch(void* const* d_in, const int* in_sizes, int n_in,
                              void* d_out, int out_size, void* d_ws, size_t ws_size,
                              hipStream_t stream)
{
    const float* queries = (const float*)d_in[0];
    const float* keys    = (const float*)d_in[1];
    const float* Wq      = (const float*)d_in[2];
    const float* bq      = (const float*)d_in[3];
    const float* Wk      = (const float*)d_in[4];
    const float* bk      = (const float*)d_in[5];
    const float* Wv      = (const float*)d_in[6];
    const float* bv      = (const float*)d_in[7];
    const int*   caus    = (const int*)d_in[8];

    float* out  = (float*)d_out;                              // [B,S,DIM]
    float* attn = out + (size_t)BB * SS * DIMM;               // [B,H,S,S]

    float* Qb    = (float*)d_ws;                              // [B*S, DIM]
    float* Kb    = Qb + (size_t)BB * SS * DIMM;
    float* Vb    = Kb + (size_t)BB * SS * DIMM;
    float* qmask = Vb + (size_t)BB * SS * DIMM;               // [B*S]
    float* kmask = qmask + (size_t)BB * SS;                   // [B*S]

    // 1) padding masks
    masks_kernel<<<dim3((2 * BB * SS) / 256), dim3(256), 0, stream>>>(
        queries, keys, qmask, kmask);

    // 2) Q/K/V projections (z = 0/1/2), 32x32 per wave
    qkv_proj_kernel<<<dim3(256, 1, 3), dim3(256), 0, stream>>>(
        queries, keys, Wq, bq, Wk, bk, Wv, bv, Qb, Kb, Vb);

    // 3) scores into attention-weights region, 64x64 per wave
    scores_kernel<<<dim3(128, 1, BB * HH), dim3(256), 0, stream>>>(
        Qb, Kb, attn, caus);

    // 4) masked softmax in place
    softmax_kernel<<<dim3(BB * HH * SS), dim3(256), 0, stream>>>(
        attn, qmask, kmask, caus);

    // 5) attn @ V -> merged-head output, 16x64 per wave
    av_kernel<<<dim3(256), dim3(256), 0, stream>>>(
        attn, Vb, out, caus);
}